// LFVSSMBlockV66_18519898981120
// MI455X (gfx1250) — compile-verified
//
#include <hip/hip_runtime.h>
#include <hip/hip_bf16.h>
#include <math.h>

// ---- problem constants ----
#define B_   4
#define C_   128
#define H_   64
#define W_   64
#define HW_  4096        // H_*W_ == L
#define L_   4096
#define G_   32          // C_/4
#define DI_  160         // D_INNER
#define DS_  24          // D_STATE
#define RK_  8           // DT_RANK
#define NX_  56          // RK_ + 2*DS_
#define XZW_ 320         // 2*DI_

typedef __attribute__((ext_vector_type(2))) float v2f;
typedef __attribute__((ext_vector_type(8))) float v8f;

// =====================================================================
// 1) Pre-LayerNorm over channel dim at each (b,h,w).  One wave per pixel,
//    each lane owns 4 channels, wave-wide shuffle reduction (wave32).
// =====================================================================
__global__ __launch_bounds__(256)
void preln_kernel(const float* __restrict__ x, const float* __restrict__ g,
                  const float* __restrict__ bt, float* __restrict__ xn)
{
    int lane = threadIdx.x & 31;
    int wave = threadIdx.x >> 5;
    long p  = (long)blockIdx.x * 8 + wave;          // pixel over B*HW
    long b  = p >> 12;
    long hw = p & (HW_ - 1);
    float v[4]; float sum = 0.f, sq = 0.f;
    #pragma unroll
    for (int i = 0; i < 4; ++i) {
        int c = lane + 32 * i;
        v[i] = x[(b * C_ + c) * (long)HW_ + hw];
        sum += v[i]; sq += v[i] * v[i];
    }
    #pragma unroll
    for (int o = 16; o > 0; o >>= 1) {
        sum += __shfl_xor(sum, o, 32);
        sq  += __shfl_xor(sq,  o, 32);
    }
    float mu  = sum * (1.f / C_);
    float var = sq * (1.f / C_) - mu * mu;
    float rs  = rsqrtf(var + 1e-5f);
    #pragma unroll
    for (int i = 0; i < 4; ++i) {
        int c = lane + 32 * i;
        xn[(b * C_ + c) * (long)HW_ + hw] = (v[i] - mu) * rs * g[c] + bt[c];
    }
}

// =====================================================================
// 2) Local branch group convs -> ybuf [B,C,HW]
//    group0: 32x32 conv1x1, groups1..3: depthwise 3x3 SAME
// =====================================================================
__global__ __launch_bounds__(256)
void groupconv_kernel(const float* __restrict__ xn,
                      const float* __restrict__ w1,
                      const float* __restrict__ dw1,
                      const float* __restrict__ dw2,
                      const float* __restrict__ dw3,
                      float* __restrict__ ybuf)
{
    long idx = (long)blockIdx.x * blockDim.x + threadIdx.x;   // B*C*HW
    if (idx >= (long)B_ * C_ * HW_) return;
    int  hw = (int)(idx & (HW_ - 1));
    long t  = idx >> 12;
    int  c  = (int)(t & (C_ - 1));
    long b  = t >> 7;
    int h = hw >> 6, w = hw & 63;
    float r;
    if (c < G_) {
        const float* xp = xn + b * C_ * (long)HW_ + hw;
        float s = 0.f;
        #pragma unroll 8
        for (int ic = 0; ic < G_; ++ic) s += w1[c * G_ + ic] * xp[(long)ic * HW_];
        r = s;
    } else {
        const float* dw = (c < 2*G_) ? dw1 + (c -   G_) * 9
                        : (c < 3*G_) ? dw2 + (c - 2*G_) * 9
                                     : dw3 + (c - 3*G_) * 9;
        const float* xp = xn + (b * C_ + c) * (long)HW_;
        float s = 0.f;
        #pragma unroll
        for (int dy = -1; dy <= 1; ++dy)
            #pragma unroll
            for (int dx = -1; dx <= 1; ++dx) {
                int hh = h + dy, ww = w + dx;
                if (hh >= 0 && hh < H_ && ww >= 0 && ww < W_)
                    s += dw[(dy + 1) * 3 + (dx + 1)] * xp[hh * W_ + ww];
            }
        r = s;
    }
    ybuf[idx] = r;
}

// =====================================================================
// 3) Generic strided f32 WMMA GEMM: one wave -> 32x32 D macro-tile
//    (2x2 grid of 16x16 WMMA tiles; 4 WMMAs per k-step, A/B fragments
//    reused across the register tile; branch-free N guard keeps EXEC
//    all-ones through the whole k-loop).
//    out(m,n) = epilogue( sum_k A(m,k)*B(k,n) )
//    A(m,k)  at A[b*a_bs + m*a_rs + k*a_cs]
//    B(k,n)  at Bw[k*b_ks + n*b_ns]            (weights, no batch)
//    out     at Out[b*o_bs + o_off + m*o_rs + n*o_cs]
//    Res(m,n)at Res[b*r_bs + m*r_rs + n*r_cs]  (optional)
//    mode: 0 plain | 1 leakyrelu(0.1)+Res | 2 Res + s*acc | 3 softplus(acc+bias[n])
//    K multiple of 4; M (per batch) multiple of 32; N guarded.
// =====================================================================
__global__ __launch_bounds__(32)
void gemm_wmma_f32(const float* __restrict__ A, long a_rs, long a_cs, long a_bs,
                   const float* __restrict__ Bw, long b_ks, long b_ns,
                   float* __restrict__ Out, long o_rs, long o_cs, long o_bs, long o_off,
                   const float* __restrict__ Res, long r_rs, long r_cs, long r_bs,
                   const float* __restrict__ bias,
                   const float* __restrict__ sptr,
                   int N, int K, int mode)
{
    const int lane = threadIdx.x;          // 0..31, full wave, EXEC all ones
    const int n0 = blockIdx.x * 32;
    const int m0 = blockIdx.y * 32;
    const int b  = blockIdx.z;
    const float* Ab = A + (long)b * a_bs;

    // ISA 7.12.2 layouts: A 16x4 -> lanes 0..15 hold K={0,1}, lanes 16..31 K={2,3}
    // B 4x16 mirrored; C/D: VGPR r = rows r (lanes 0..15) and r+8 (lanes 16..31).
    const int row = lane & 15;
    const int kk2 = (lane >> 4) * 2;       // 0 or 2

    // branch-free N guard: clamp OOB columns to 0 and zero the fragment
    const int   nb0  = n0 + row;
    const int   nb1  = n0 + 16 + row;
    const float msk0 = (nb0 < N) ? 1.f : 0.f;
    const float msk1 = (nb1 < N) ? 1.f : 0.f;
    const long  nc0  = (nb0 < N) ? nb0 : 0;
    const long  nc1  = (nb1 < N) ? nb1 : 0;

    v8f acc00 = {}, acc01 = {}, acc10 = {}, acc11 = {};
    for (int k0 = 0; k0 < K; k0 += 4) {
        v2f a0, a1, b0, b1;
        long ab0 = (long)(m0 + row) * a_rs + (long)(k0 + kk2) * a_cs;
        long ab1 = ab0 + 16 * a_rs;
        a0.x = Ab[ab0];        a0.y = Ab[ab0 + a_cs];
        a1.x = Ab[ab1];        a1.y = Ab[ab1 + a_cs];
        long kb  = (long)(k0 + kk2) * b_ks;
        long bb0 = kb + nc0 * b_ns;
        long bb1 = kb + nc1 * b_ns;
        b0.x = Bw[bb0] * msk0; b0.y = Bw[bb0 + b_ks] * msk0;
        b1.x = Bw[bb1] * msk1; b1.y = Bw[bb1 + b_ks] * msk1;
        acc00 = __builtin_amdgcn_wmma_f32_16x16x4_f32(false, a0, false, b0, (short)0, acc00, false, false);
        acc01 = __builtin_amdgcn_wmma_f32_16x16x4_f32(false, a0, false, b1, (short)0, acc01, false, false);
        acc10 = __builtin_amdgcn_wmma_f32_16x16x4_f32(false, a1, false, b0, (short)0, acc10, false, false);
        acc11 = __builtin_amdgcn_wmma_f32_16x16x4_f32(false, a1, false, b1, (short)0, acc11, false, false);
    }

    // epilogue + store
    const int col  = lane & 15;
    const int half = lane >> 4;
    float* Ob = Out + (long)b * o_bs + o_off;
    const float* Rb = Res ? (Res + (long)b * r_bs) : nullptr;
    const float s = sptr ? sptr[0] : 1.0f;

    auto epi = [&](float v, int m, int n) -> float {
        if (mode == 1) {
            float lv = v > 0.f ? v : 0.1f * v;
            return lv + Rb[(long)m * r_rs + (long)n * r_cs];
        } else if (mode == 2) {
            return Rb[(long)m * r_rs + (long)n * r_cs] + s * v;
        } else if (mode == 3) {
            float tt = v + bias[n];
            return (tt > 20.f) ? tt : log1pf(__expf(tt));
        }
        return v;
    };

    const int nA = n0 + col;
    const int nB = nA + 16;
    #pragma unroll
    for (int r = 0; r < 8; ++r) {
        int mA = m0 + r + half * 8;
        int mB = mA + 16;
        if (nA < N) {
            Ob[(long)mA * o_rs + (long)nA * o_cs] = epi(acc00[r], mA, nA);
            Ob[(long)mB * o_rs + (long)nA * o_cs] = epi(acc10[r], mB, nA);
        }
        if (nB < N) {
            Ob[(long)mA * o_rs + (long)nB * o_cs] = epi(acc01[r], mA, nB);
            Ob[(long)mB * o_rs + (long)nB * o_cs] = epi(acc11[r], mB, nB);
        }
    }
}

// =====================================================================
// 4) Build scan sequence (4 scan orders) + LayerNorm -> seq [B,L,C] row-major
// =====================================================================
__global__ __launch_bounds__(256)
void seqln_kernel(const float* __restrict__ xn, const float* __restrict__ g,
                  const float* __restrict__ bt, float* __restrict__ seq)
{
    int lane = threadIdx.x & 31;
    int wave = threadIdx.x >> 5;
    long p = (long)blockIdx.x * 8 + wave;     // (b,l)
    long b = p >> 12;
    int  l = (int)(p & (L_ - 1));
    int lr = L_ - 1 - l;
    int hw0 = l;
    int hw1 = lr;
    int hw2 = ((l  & 63) << 6) | (l  >> 6);   // transpose scan
    int hw3 = ((lr & 63) << 6) | (lr >> 6);   // reversed transpose scan
    const float* xb = xn + b * C_ * (long)HW_;
    float v0 = xb[(long)(lane      ) * HW_ + hw0];
    float v1 = xb[(long)(lane + 32 ) * HW_ + hw1];
    float v2 = xb[(long)(lane + 64 ) * HW_ + hw2];
    float v3 = xb[(long)(lane + 96 ) * HW_ + hw3];
    float sum = v0 + v1 + v2 + v3;
    float sq  = v0*v0 + v1*v1 + v2*v2 + v3*v3;
    #pragma unroll
    for (int o = 16; o > 0; o >>= 1) {
        sum += __shfl_xor(sum, o, 32);
        sq  += __shfl_xor(sq,  o, 32);
    }
    float mu  = sum * (1.f / C_);
    float var = sq * (1.f / C_) - mu * mu;
    float rs  = rsqrtf(var + 1e-5f);
    float* sp = seq + (b * L_ + l) * (long)C_;
    sp[lane      ] = (v0 - mu) * rs * g[lane      ] + bt[lane      ];
    sp[lane + 32 ] = (v1 - mu) * rs * g[lane + 32 ] + bt[lane + 32 ];
    sp[lane + 64 ] = (v2 - mu) * rs * g[lane + 64 ] + bt[lane + 64 ];
    sp[lane + 96 ] = (v3 - mu) * rs * g[lane + 96 ] + bt[lane + 96 ];
}

// =====================================================================
// 5) Causal depthwise conv1d (width 4) + SiLU over xz[...,:DI] -> xr [B,L,DI]
// =====================================================================
__global__ __launch_bounds__(256)
void conv1d_silu_kernel(const float* __restrict__ xz, const float* __restrict__ cw,
                        const float* __restrict__ cb, float* __restrict__ xr)
{
    long idx = (long)blockIdx.x * blockDim.x + threadIdx.x;  // B*L*DI
    if (idx >= (long)B_ * L_ * DI_) return;
    int  d = (int)(idx % DI_);
    long t = idx / DI_;
    int  l = (int)(t & (L_ - 1));
    long b = t >> 12;
    float s = cb[d];
    #pragma unroll
    for (int j = 0; j < 4; ++j) {
        int ls = l - 3 + j;
        if (ls >= 0) s += cw[d * 4 + j] * xz[((b << 12) + ls) * (long)XZW_ + d];
    }
    xr[idx] = s / (1.f + __expf(-s));       // silu
}

// =====================================================================
// 6) Selective scan. Block = one batch (160 threads = channels); each thread
//    carries the 24-wide state in registers; B/C projections staged via LDS
//    in 32-step chunks.  Fuses +xr*D and *silu(z) gate; ygate may alias delta.
// =====================================================================
__global__ __launch_bounds__(160)
void scan_kernel(const float* __restrict__ delta,  // [B,L,DI]
                 const float* __restrict__ xr,     // [B,L,DI]
                 const float* __restrict__ dbl,    // [B,L,NX]  (B at 8, C at 32)
                 const float* __restrict__ xz,     // [B,L,2*DI] (z at DI)
                 const float* __restrict__ A_log,  // [DI,DS]
                 const float* __restrict__ Dp,     // [DI]
                 float* __restrict__ ygate)        // [B,L,DI]
{
    const int d = threadIdx.x;
    const int b = blockIdx.x;
    __shared__ float sBC[32 * 48];
    float a[DS_], h[DS_];
    #pragma unroll
    for (int n = 0; n < DS_; ++n) {
        a[n] = -__expf(A_log[d * DS_ + n]);
        h[n] = 0.f;
    }
    const float Dd = Dp[d];
    for (int l0 = 0; l0 < L_; l0 += 32) {
        __syncthreads();
        for (int i = threadIdx.x; i < 32 * 48; i += blockDim.x) {
            int li = i / 48, ci = i % 48;
            sBC[i] = dbl[((long)b * L_ + l0 + li) * NX_ + RK_ + ci];
        }
        __syncthreads();
        for (int li = 0; li < 32; ++li) {
            long idx = ((long)b * L_ + l0 + li) * (long)DI_ + d;
            float dl = delta[idx];
            float xv = xr[idx];
            float zv = xz[((long)b * L_ + l0 + li) * (long)XZW_ + DI_ + d];
            const float* Bm = &sBC[li * 48];
            const float* Cm = Bm + DS_;
            float y = 0.f;
            #pragma unroll
            for (int n = 0; n < DS_; ++n) {
                float dA = __expf(dl * a[n]);
                h[n] = dA * h[n] + (dl * Bm[n]) * xv;
                y += h[n] * Cm[n];
            }
            y += xv * Dd;
            float sz = zv / (1.f + __expf(-zv));
            ygate[idx] = y * sz;
        }
    }
}

// =====================================================================
// 7) Inverse scan rearrange: out_seq [B,L,C] -> rseq [B,HW,C] row-major
// =====================================================================
__global__ __launch_bounds__(256)
void rearrange_kernel(const float* __restrict__ os, float* __restrict__ rseq)
{
    long idx = (long)blockIdx.x * blockDim.x + threadIdx.x;  // B*HW*C
    if (idx >= (long)B_ * HW_ * C_) return;
    int  c  = (int)(idx & (C_ - 1));
    long t  = idx >> 7;
    int  hw = (int)(t & (HW_ - 1));
    long b  = t >> 12;
    int h = hw >> 6, w = hw & 63;
    int l;
    if      (c <  G_)     l = hw;
    else if (c < 2*G_)    l = L_ - 1 - hw;
    else if (c < 3*G_)    l = (w << 6) | h;
    else                  l = L_ - 1 - ((w << 6) | h);
    rseq[idx] = os[((b << 12) + l) * (long)C_ + c];
}

// =====================================================================
// 8) Global average pool over HW per (b,c)
// =====================================================================
__global__ __launch_bounds__(256)
void pool_kernel(const float* __restrict__ fused, float* __restrict__ pooled)
{
    int bc = blockIdx.x;                    // 0..B*C-1
    const float* p = fused + (long)bc * HW_;
    float s = 0.f;
    for (int i = threadIdx.x; i < HW_; i += 256) s += p[i];
    __shared__ float red[256];
    red[threadIdx.x] = s; __syncthreads();
    #pragma unroll
    for (int o = 128; o > 0; o >>= 1) {
        if (threadIdx.x < o) red[threadIdx.x] += red[threadIdx.x + o];
        __syncthreads();
    }
    if (threadIdx.x == 0) pooled[bc] = red[0] * (1.f / HW_);
}

// =====================================================================
// 9) Channel-attention MLP: pooled[B,128] -> relu(fc1) -> sigmoid(fc2)
// =====================================================================
__global__ __launch_bounds__(128)
void att_kernel(const float* __restrict__ pooled,
                const float* __restrict__ w1, const float* __restrict__ b1,
                const float* __restrict__ w2, const float* __restrict__ b2,
                float* __restrict__ attn)
{
    __shared__ float hv[B_ * 16];
    int t = threadIdx.x;
    if (t < B_ * 16) {
        int b = t >> 4, j = t & 15;
        float s = b1[j];
        for (int c = 0; c < C_; ++c) s += w1[j * C_ + c] * pooled[b * C_ + c];
        hv[t] = s > 0.f ? s : 0.f;
    }
    __syncthreads();
    int c = t;
    for (int b = 0; b < B_; ++b) {
        float s = b2[c];
        #pragma unroll
        for (int j = 0; j < 16; ++j) s += w2[c * 16 + j] * hv[b * 16 + j];
        attn[b * C_ + c] = 1.f / (1.f + __expf(-s));
    }
}

// =====================================================================
// 10) Final residual: out = x + res_scale * fused * attn[b,c]
// =====================================================================
__global__ __launch_bounds__(256)
void final_kernel(const float* __restrict__ x, const float* __restrict__ fused,
                  const float* __restrict__ attn, const float* __restrict__ rsc,
                  float* __restrict__ out)
{
    long idx = (long)blockIdx.x * blockDim.x + threadIdx.x;  // B*C*HW
    if (idx >= (long)B_ * C_ * HW_) return;
    long t = idx >> 12;
    int  c = (int)(t & (C_ - 1));
    long b = t >> 7;
    out[idx] = x[idx] + rsc[0] * fused[idx] * attn[b * C_ + c];
}

// =====================================================================
// Launch
// =====================================================================
extern "C" void kernel_launch(void* const* d_in, const int* in_sizes, int n_in,
                              void* d_out, int out_size, void* d_ws, size_t ws_size,
                              hipStream_t stream)
{
    const float* x          = (const float*)d_in[0];
    const float* pre_gamma  = (const float*)d_in[1];
    const float* pre_beta   = (const float*)d_in[2];
    const float* lb_conv1_w = (const float*)d_in[3];
    const float* lb_dw1_w   = (const float*)d_in[4];
    const float* lb_dw2_w   = (const float*)d_in[5];
    const float* lb_dw3_w   = (const float*)d_in[6];
    const float* lb_pw_w    = (const float*)d_in[7];
    const float* gb_gamma   = (const float*)d_in[8];
    const float* gb_beta    = (const float*)d_in[9];
    const float* gb_fus_w   = (const float*)d_in[10];
    const float* gb_scale   = (const float*)d_in[11];
    const float* m_in_w     = (const float*)d_in[12];
    const float* m_conv_w   = (const float*)d_in[13];
    const float* m_conv_b   = (const float*)d_in[14];
    const float* m_xproj_w  = (const float*)d_in[15];
    const float* m_dt_w     = (const float*)d_in[16];
    const float* m_dt_b     = (const float*)d_in[17];
    const float* m_A_log    = (const float*)d_in[18];
    const float* m_D        = (const float*)d_in[19];
    const float* m_out_w    = (const float*)d_in[20];
    const float* fuse_w     = (const float*)d_in[21];
    const float* fc1_w      = (const float*)d_in[22];
    const float* fc1_b      = (const float*)d_in[23];
    const float* fc2_w      = (const float*)d_in[24];
    const float* fc2_b      = (const float*)d_in[25];
    const float* res_scale  = (const float*)d_in[26];
    float* out = (float*)d_out;

    float* ws = (float*)d_ws;
    // workspace layout (floats) -- total ~24.0M floats (~96 MB)
    const long SZ_BCHW = (long)B_ * C_ * HW_;          // 2,097,152
    float* XN    = ws;                                  // [B,C,HW]
    float* YB    = XN    + SZ_BCHW;                     // ybuf / out_seq
    float* SEQ   = YB    + SZ_BCHW;                     // seq / rseq
    float* CAT   = SEQ   + SZ_BCHW;                     // [B,HW,256]
    float* XZ    = CAT   + (long)B_ * HW_ * 2 * C_;     // [B,L,320]
    float* XR    = XZ    + (long)B_ * L_ * XZW_;        // [B,L,160]
    float* DBL   = XR    + (long)B_ * L_ * DI_;         // [B,L,56]
    float* DELTA = DBL   + (long)B_ * L_ * NX_;         // [B,L,160] (also ygate)
    float* FUSED = DELTA + (long)B_ * L_ * DI_;         // [B,C,HW]
    float* POOL  = FUSED + SZ_BCHW;                     // [B*C]
    float* ATTN  = POOL  + B_ * C_;                     // [B*C]

    const int MT = HW_ / 32;   // 128 macro-tiles of 32 rows per batch

    // 1) pre-LN
    preln_kernel<<<dim3((B_ * HW_) / 8), dim3(256), 0, stream>>>(x, pre_gamma, pre_beta, XN);

    // 2) local group convs
    groupconv_kernel<<<dim3((unsigned)(SZ_BCHW / 256)), dim3(256), 0, stream>>>(
        XN, lb_conv1_w, lb_dw1_w, lb_dw2_w, lb_dw3_w, YB);

    // 3) pw conv1x1 GEMM, epilogue leaky(0.1)+xn -> CAT[:, :, 0:128]
    gemm_wmma_f32<<<dim3(C_/32, MT, B_), dim3(32), 0, stream>>>(
        YB, 1, HW_, (long)C_ * HW_,
        lb_pw_w, 1, C_,
        CAT, 2 * C_, 1, (long)HW_ * 2 * C_, 0,
        XN, 1, HW_, (long)C_ * HW_,
        nullptr, nullptr, C_, C_, 1);

    // 4) scan-sequence gather + LN
    seqln_kernel<<<dim3((B_ * L_) / 8), dim3(256), 0, stream>>>(XN, gb_gamma, gb_beta, SEQ);

    // 5) in_proj GEMM: seq[B,L,128] x W[320,128]^T -> XZ[B,L,320]
    gemm_wmma_f32<<<dim3(XZW_/32, MT, B_), dim3(32), 0, stream>>>(
        SEQ, C_, 1, (long)L_ * C_,
        m_in_w, 1, C_,
        XZ, XZW_, 1, (long)L_ * XZW_, 0,
        nullptr, 0, 0, 0, nullptr, nullptr, XZW_, C_, 0);

    // 6) causal conv1d + silu -> XR
    conv1d_silu_kernel<<<dim3((unsigned)(((long)B_*L_*DI_) / 256)), dim3(256), 0, stream>>>(
        XZ, m_conv_w, m_conv_b, XR);

    // 7) x_proj GEMM: XR x W[56,160]^T -> DBL (N=56, masked)
    gemm_wmma_f32<<<dim3((NX_ + 31)/32, MT, B_), dim3(32), 0, stream>>>(
        XR, DI_, 1, (long)L_ * DI_,
        m_xproj_w, 1, DI_,
        DBL, NX_, 1, (long)L_ * NX_, 0,
        nullptr, 0, 0, 0, nullptr, nullptr, NX_, DI_, 0);

    // 8) dt GEMM: DBL[:,:, :8] x dt_w[160,8]^T, softplus(+bias) -> DELTA
    gemm_wmma_f32<<<dim3(DI_/32, MT, B_), dim3(32), 0, stream>>>(
        DBL, NX_, 1, (long)L_ * NX_,
        m_dt_w, 1, RK_,
        DELTA, DI_, 1, (long)L_ * DI_, 0,
        nullptr, 0, 0, 0,
        m_dt_b, nullptr, DI_, RK_, 3);

    // 9) selective scan (fuses +xr*D and silu(z) gate); ygate aliases DELTA
    scan_kernel<<<dim3(B_), dim3(DI_), 0, stream>>>(
        DELTA, XR, DBL, XZ, m_A_log, m_D, DELTA);

    // 10) out_proj GEMM: ygate x W[128,160]^T -> out_seq (reuse YB)
    gemm_wmma_f32<<<dim3(C_/32, MT, B_), dim3(32), 0, stream>>>(
        DELTA, DI_, 1, (long)L_ * DI_,
        m_out_w, 1, DI_,
        YB, C_, 1, (long)L_ * C_, 0,
        nullptr, 0, 0, 0, nullptr, nullptr, C_, DI_, 0);

    // 11) inverse-scan rearrange -> rseq (reuse SEQ)
    rearrange_kernel<<<dim3((unsigned)(SZ_BCHW / 256)), dim3(256), 0, stream>>>(YB, SEQ);

    // 12) gb fusion GEMM: glob = xn + gb_scale*conv -> CAT[:, :, 128:256]
    gemm_wmma_f32<<<dim3(C_/32, MT, B_), dim3(32), 0, stream>>>(
        SEQ, C_, 1, (long)L_ * C_,
        gb_fus_w, 1, C_,
        CAT, 2 * C_, 1, (long)HW_ * 2 * C_, C_,
        XN, 1, HW_, (long)C_ * HW_,
        nullptr, gb_scale, C_, C_, 2);

    // 13) fuse GEMM: CAT[B,HW,256] x fuse_w[128,256]^T -> FUSED [B,C,HW]
    gemm_wmma_f32<<<dim3(C_/32, MT, B_), dim3(32), 0, stream>>>(
        CAT, 2 * C_, 1, (long)HW_ * 2 * C_,
        fuse_w, 1, 2 * C_,
        FUSED, 1, HW_, (long)C_ * HW_, 0,
        nullptr, 0, 0, 0, nullptr, nullptr, C_, 2 * C_, 0);

    // 14) global average pool
    pool_kernel<<<dim3(B_ * C_), dim3(256), 0, stream>>>(FUSED, POOL);

    // 15) channel attention MLP
    att_kernel<<<dim3(1), dim3(C_), 0, stream>>>(POOL, fc1_w, fc1_b, fc2_w, fc2_b, ATTN);

    // 16) final residual
    final_kernel<<<dim3((unsigned)(SZ_BCHW / 256)), dim3(256), 0, stream>>>(
        x, FUSED, ATTN, res_scale, out);
}